// ViterbiLoss_82420422410341
// MI455X (gfx1250) — compile-verified
//
#include <hip/hip_runtime.h>
#include <hip/hip_bf16.h>
#include <math.h>

// Problem constants (from reference)
#define TAGS 64
#define START_TAG 62
#define STOP_TAG 63
#define BB 32
#define SS 256
#define MAT (TAGS * TAGS) // 4096 floats = 16KB

typedef float v2f __attribute__((ext_vector_type(2)));
typedef float v8f __attribute__((ext_vector_type(8)));

// ---------------------------------------------------------------------------
// One 64x64x64 f32 matmul step entirely in LDS, using V_WMMA_F32_16X16X4_F32.
//   C = A x Bm            (A, Bm, C: 64x64 row-major in LDS)
// (Any running rescale is pre-folded into Bm during staging, so the inner
//  loop is pure ds_load + v_wmma.)
// 8 waves (256 threads); each wave computes 2 of the 16 16x16 output tiles.
// Also produces per-tile max into tileMax[16] (entries >= 0: products of
// exponentials), for the running rescale of the matrix-product scan.
//
// VGPR layouts per CDNA5 ISA (CDNA5 ISA):
//   A 16x4 f32 : lanes 0-15 -> M=lane, {K=0,K=1}; lanes 16-31 -> {K=2,K=3}
//   B 4x16 f32 : mirrored (lanes 0-15 -> N=lane, rows K=0/1; 16-31 -> K=2/3)
//   C/D 16x16  : VGPR r -> (M=r, N=lane) lanes 0-15 ; (M=r+8, N=lane-16)
// ---------------------------------------------------------------------------
__device__ __forceinline__ void mm64_step(const float* __restrict__ A,
                                          const float* __restrict__ Bm,
                                          float* __restrict__ C,
                                          float* __restrict__ tileMax) {
  const int lane = threadIdx.x & 31;
  const int wave = threadIdx.x >> 5;
  const int half = lane >> 4;   // 0: lanes 0-15, 1: lanes 16-31
  const int l    = lane & 15;

  for (int tile = wave; tile < 16; tile += 8) {   // uniform per wave: 2 tiles
    const int mi = tile >> 2;
    const int ni = tile & 3;
    v8f acc = {0.f, 0.f, 0.f, 0.f, 0.f, 0.f, 0.f, 0.f};
    const int arow = mi * 16 + l;
    const int bcol = ni * 16 + l;
#pragma unroll
    for (int kk = 0; kk < 16; ++kk) {             // K = 64 in chunks of 4
      const int k0 = kk * 4 + 2 * half;
      v2f a, b;
      a.x = A[arow * 64 + k0];
      a.y = A[arow * 64 + k0 + 1];
      b.x = Bm[(k0)     * 64 + bcol];
      b.y = Bm[(k0 + 1) * 64 + bcol];
      acc = __builtin_amdgcn_wmma_f32_16x16x4_f32(false, a, false, b,
                                                  (short)0, acc, false, false);
    }
    float tm = 0.f;
#pragma unroll
    for (int r = 0; r < 8; ++r) {
      const int row = mi * 16 + r + (half ? 8 : 0);
      C[row * 64 + (ni * 16 + l)] = acc[r];
      tm = fmaxf(tm, acc[r]);
    }
    // deterministic wave32 max-reduce
    for (int off = 16; off > 0; off >>= 1)
      tm = fmaxf(tm, __shfl_xor(tm, off, 32));
    if (lane == 0) tileMax[tile] = tm;
  }
}

// ---------------------------------------------------------------------------
// Gold-path score: sum over valid (b,s) of features[b,s].flat[targets[b,s]].
// Single WG, fixed-order tree reduction -> deterministic.
// ---------------------------------------------------------------------------
__global__ void __launch_bounds__(256)
crf_gold_kernel(const float* __restrict__ feats, const int* __restrict__ targets,
                const int* __restrict__ lengths, float* __restrict__ gold) {
  __shared__ float red[256];
  const int tid = threadIdx.x;
  float s = 0.f;
  for (int i = tid; i < BB * SS; i += 256) {
    const int b  = i >> 8;     // S == 256
    const int ss = i & 255;
    if (ss < lengths[b])
      s += feats[((size_t)i << 12) + targets[i]];
  }
  red[tid] = s;
  __syncthreads();
  for (int off = 128; off > 0; off >>= 1) {
    if (tid < off) red[tid] += red[tid + off];
    __syncthreads();
  }
  if (tid == 0) gold[0] = red[0];
}

// ---------------------------------------------------------------------------
// Chunked parallel scan, stage 1: each WG (b, c) computes the normalized
// product of its chunk's transition matrices in exp space:
//   T_chunk = exp(logscale) * P_norm,  P_norm entries <= 1.
// Software-pipelined: global loads for step t+1 are issued before the step-t
// WMMAs (latency hidden under the matmul); exp + rescale + LDS store are
// deferred until after the max-update barrier so the staged tile is scaled
// by the fresh 1/m_t. Per-step adaptive rescale keeps f32 in range for any
// chunk length.
// ---------------------------------------------------------------------------
__global__ void __launch_bounds__(256)
crf_chunk_kernel(const float* __restrict__ feats, const int* __restrict__ lengths,
                 float* __restrict__ chunkMat, float* __restrict__ logscale,
                 int nchunk, int chunk_len) {
  __shared__ float P[2][MAT];   // ping-pong product (32KB)
  __shared__ float E[2][MAT];   // double-buffered exp(features) (32KB)
  __shared__ float tileMax[16];
  __shared__ float st[3];       // 0: m_prev, 1: L (log-scale), 2: 1/m_prev

  const int b   = blockIdx.x;
  const int c   = blockIdx.y;
  const int tid = threadIdx.x;
  const int len = lengths[b];

  // P[0] = identity
  for (int k = 0; k < 16; ++k) {
    const int idx = k * 256 + tid;
    P[0][idx] = ((idx >> 6) == (idx & 63)) ? 1.f : 0.f;
  }
  if (tid == 0) { st[0] = 1.f; st[1] = 0.f; st[2] = 1.f; }

  const int t0   = 1 + c * chunk_len;
  const int tEnd = min(min(t0 + chunk_len, SS), len);  // steps: t in [t0,tEnd)

  // Prologue: stage E[0] = exp(features[b,t0,:,:]) (scale 1/m_{-1} = 1).
  if (t0 < tEnd) {
    const float4* src = reinterpret_cast<const float4*>(
        feats + (((size_t)(b * SS + t0)) << 12));
    float4* dst = reinterpret_cast<float4*>(E[0]);
#pragma unroll
    for (int k = 0; k < 4; ++k) {
      float4 v = src[k * 256 + tid];
      v.x = expf(v.x); v.y = expf(v.y); v.z = expf(v.z); v.w = expf(v.w);
      dst[k * 256 + tid] = v;
    }
  }
  __syncthreads();

  int cur = 0, ce = 0;
  for (int t = t0; t < tEnd; ++t) {
    // Issue next step's global loads now; latency hides under the WMMAs.
    const bool haveNext = (t + 1 < tEnd);
    float4 nxt[4];
    if (haveNext) {
      const float4* src = reinterpret_cast<const float4*>(
          feats + (((size_t)(b * SS + t + 1)) << 12));
#pragma unroll
      for (int k = 0; k < 4; ++k) nxt[k] = src[k * 256 + tid];
    }

    mm64_step(P[cur], E[ce], P[cur ^ 1], tileMax);
    __syncthreads();

    if (tid == 0) {
      float m = tileMax[0];
      for (int i = 1; i < 16; ++i) m = fmaxf(m, tileMax[i]);
      st[1] += logf(st[0]);   // consume previous normalizer
      st[0] = m;
      st[2] = 1.f / m;
    }
    __syncthreads();

    // Stage E for t+1, pre-scaled by the fresh 1/m_t:  P x (E/m) == (P/m) x E
    if (haveNext) {
      const float inv = st[2];
      float4* dst = reinterpret_cast<float4*>(E[ce ^ 1]);
#pragma unroll
      for (int k = 0; k < 4; ++k) {
        float4 v = nxt[k];
        v.x = expf(v.x) * inv; v.y = expf(v.y) * inv;
        v.z = expf(v.z) * inv; v.w = expf(v.w) * inv;
        dst[k * 256 + tid] = v;
      }
    }
    cur ^= 1; ce ^= 1;
    __syncthreads();
  }

  // Emit P_norm = P / m_last and logscale = L + log(m_last).
  const float inv = st[2];
  float* outM = chunkMat + ((size_t)(b * nchunk + c)) * MAT;
  for (int k = 0; k < 16; ++k) {
    const int idx = k * 256 + tid;
    outM[idx] = P[cur][idx] * inv;
  }
  if (tid == 0) logscale[b * nchunk + c] = st[1] + logf(st[0]);
}

// ---------------------------------------------------------------------------
// Stage 2: per batch, combine chunk products left-to-right (same WMMA
// machinery; the running 1/m is folded into the staged B tile), then fold in
// the init vector and read column STOP_TAG:
//   score[b] = m0 + L + log( sum_i exp(v0[i]-m0) * P[i, STOP] )
// ---------------------------------------------------------------------------
__global__ void __launch_bounds__(256)
crf_combine_kernel(const float* __restrict__ feats,
                   const float* __restrict__ chunkMat,
                   const float* __restrict__ logscale,
                   float* __restrict__ scores, int nchunk) {
  __shared__ float P[2][MAT];
  __shared__ float E[MAT];
  __shared__ float tileMax[16];
  __shared__ float st[3];

  const int b   = blockIdx.x;
  const int tid = threadIdx.x;

  { // P = chunk 0
    const float4* src = reinterpret_cast<const float4*>(
        chunkMat + (size_t)b * nchunk * MAT);
    float4* dst = reinterpret_cast<float4*>(P[0]);
#pragma unroll
    for (int k = 0; k < 4; ++k) dst[k * 256 + tid] = src[k * 256 + tid];
  }
  if (tid == 0) { st[0] = 1.f; st[1] = logscale[b * nchunk]; st[2] = 1.f; }
  __syncthreads();

  int cur = 0;
  for (int c = 1; c < nchunk; ++c) {
    // Stage chunk c, pre-scaled by 1/m_prev (valid: st[2] settled by the
    // barrier that ended the previous iteration).
    const float inv = st[2];
    const float4* src = reinterpret_cast<const float4*>(
        chunkMat + ((size_t)(b * nchunk + c)) * MAT);
    float4* dst = reinterpret_cast<float4*>(E);
#pragma unroll
    for (int k = 0; k < 4; ++k) {
      float4 v = src[k * 256 + tid];
      v.x *= inv; v.y *= inv; v.z *= inv; v.w *= inv;
      dst[k * 256 + tid] = v;
    }
    __syncthreads();

    mm64_step(P[cur], E, P[cur ^ 1], tileMax);
    __syncthreads();

    if (tid == 0) {
      float m = tileMax[0];
      for (int i = 1; i < 16; ++i) m = fmaxf(m, tileMax[i]);
      st[1] += logf(st[0]) + logscale[b * nchunk + c];
      st[0] = m;
      st[2] = 1.f / m;
    }
    cur ^= 1;
    __syncthreads();
  }

  if (tid == 0) {
    // v0[i] = features[b, 0, START_TAG, i]
    const float* v0 = feats + (((size_t)(b * SS)) << 12) + START_TAG * 64;
    float m0 = v0[0];
    for (int i = 1; i < 64; ++i) m0 = fmaxf(m0, v0[i]);
    float dot = 0.f;
    for (int i = 0; i < 64; ++i)
      dot += expf(v0[i] - m0) * P[cur][i * 64 + STOP_TAG];
    scores[b] = m0 + st[1] + logf(dot);   // st[1] excludes the last (unconsumed) max
  }
}

// ---------------------------------------------------------------------------
// Final deterministic scalar: (sum_b score[b] - gold) / B
// ---------------------------------------------------------------------------
__global__ void crf_final_kernel(const float* __restrict__ gold,
                                 const float* __restrict__ scores,
                                 float* __restrict__ out) {
  if (threadIdx.x == 0 && blockIdx.x == 0) {
    float s = 0.f;
    for (int b = 0; b < BB; ++b) s += scores[b];
    out[0] = (s - gold[0]) / (float)BB;
  }
}

extern "C" void kernel_launch(void* const* d_in, const int* in_sizes, int n_in,
                              void* d_out, int out_size, void* d_ws, size_t ws_size,
                              hipStream_t stream) {
  const float* feats   = (const float*)d_in[0];
  const int*   targets = (const int*)d_in[1];
  const int*   lengths = (const int*)d_in[2];
  float* out = (float*)d_out;
  float* ws  = (float*)d_ws;

  // Workspace layout (floats):
  //   [0]              gold score
  //   [32 .. 64)       per-batch scores
  //   [64 .. 64+B*16)  chunk log-scales
  //   [2048 .. )       chunk matrices: B * nchunk * 4096 floats
  const size_t ws_floats = ws_size / sizeof(float);
  int nchunk = 16;
  while (nchunk > 1 &&
         (size_t)2048 + (size_t)BB * (size_t)nchunk * (size_t)MAT > ws_floats)
    --nchunk;
  const int chunk_len = (SS - 1 + nchunk - 1) / nchunk;

  float* gold     = ws;
  float* scores   = ws + 32;
  float* logscale = ws + 64;
  float* chunkMat = ws + 2048;

  crf_gold_kernel<<<1, 256, 0, stream>>>(feats, targets, lengths, gold);

  dim3 grid(BB, nchunk);
  crf_chunk_kernel<<<grid, 256, 0, stream>>>(feats, lengths, chunkMat, logscale,
                                             nchunk, chunk_len);

  crf_combine_kernel<<<BB, 256, 0, stream>>>(feats, chunkMat, logscale, scores,
                                             nchunk);

  crf_final_kernel<<<1, 64, 0, stream>>>(gold, scores, out);
}